// TopKSparseAutoencoder_85289460564005
// MI455X (gfx1250) — compile-verified
//
#include <hip/hip_runtime.h>
#include <hip/hip_bf16.h>

typedef float v2f __attribute__((ext_vector_type(2)));
typedef float v8f __attribute__((ext_vector_type(8)));

#define D_MODEL 1024
#define N_LAT   16384
#define TOPK    32
#define SAE_EPS 1e-6f
#define NEG_INF (-3.402823466e+38f)

// -------------------------------------------------------------------------
// Kernel 1: per-token preprocess.  mean, std(ddof=1), and
// xc[t][d] = (x - mean)/(std+eps) - b_pre[d]   (encoder input, fp32)
// -------------------------------------------------------------------------
__global__ __launch_bounds__(256) void prep_kernel(
    const float* __restrict__ x, const float* __restrict__ b_pre,
    float* __restrict__ xc, float* __restrict__ meanArr, float* __restrict__ normArr)
{
    __shared__ float s_sum[256];
    __shared__ float s_sq[256];
    const int tok = blockIdx.x;
    const float* xr = x + (size_t)tok * D_MODEL;
    float s = 0.f, q = 0.f;
    for (int d = threadIdx.x; d < D_MODEL; d += 256) { float v = xr[d]; s += v; q += v * v; }
    s_sum[threadIdx.x] = s; s_sq[threadIdx.x] = q;
    __syncthreads();
    for (int off = 128; off > 0; off >>= 1) {
        if (threadIdx.x < off) {
            s_sum[threadIdx.x] += s_sum[threadIdx.x + off];
            s_sq[threadIdx.x]  += s_sq[threadIdx.x + off];
        }
        __syncthreads();
    }
    const float mean = s_sum[0] * (1.0f / D_MODEL);
    float var = (s_sq[0] - (float)D_MODEL * mean * mean) * (1.0f / (D_MODEL - 1));
    var = fmaxf(var, 0.f);
    const float norm = sqrtf(var) + SAE_EPS;
    if (threadIdx.x == 0) { meanArr[tok] = mean; normArr[tok] = norm; }
    const float inv = 1.0f / norm;
    float* xcr = xc + (size_t)tok * D_MODEL;
    for (int d = threadIdx.x; d < D_MODEL; d += 256)
        xcr[d] = (xr[d] - mean) * inv - b_pre[d];
}

// -------------------------------------------------------------------------
// Kernel 2: transpose W_dec [1024][16384] -> WdT [16384][1024] so the sparse
// decode gather reads contiguous rows (coalesced from L2).
// -------------------------------------------------------------------------
__global__ __launch_bounds__(256) void transpose_kernel(
    const float* __restrict__ Wdec, float* __restrict__ WdT)
{
    __shared__ float tile[32][33];
    const int bx = blockIdx.x;          // latent-block
    const int by = blockIdx.y;          // d-block
    const int tx = threadIdx.x;         // 0..31
    const int ty = threadIdx.y;         // 0..7
#pragma unroll
    for (int j = 0; j < 32; j += 8)
        tile[ty + j][tx] = Wdec[(size_t)(by * 32 + ty + j) * N_LAT + bx * 32 + tx];
    __syncthreads();
#pragma unroll
    for (int j = 0; j < 32; j += 8)
        WdT[(size_t)(bx * 32 + ty + j) * D_MODEL + by * 32 + tx] = tile[tx][ty + j];
}

// -------------------------------------------------------------------------
// Kernel 3: fused encode (fp32 WMMA) -> running top-32 -> sparse decode.
// 1 workgroup = 32 tokens (2 M-tiles), 8 waves.  Each wave owns 128 of the
// 1024 latent tiles; every B (W_enc) register load feeds TWO WMMAs (c0,c1),
// halving W_enc L2 traffic vs a single-M-tile scheme.  All 32 lanes maintain
// a per-token top-32 list (lane = token) in LDS.
// -------------------------------------------------------------------------
__global__ __launch_bounds__(256) void fused_kernel(
    const float* __restrict__ xc,   const float* __restrict__ W_enc,
    const float* __restrict__ b_enc,const float* __restrict__ b_pre,
    const float* __restrict__ WdT,  const float* __restrict__ meanArr,
    const float* __restrict__ normArr, float* __restrict__ out)
{
    __shared__ float Alds[32][1028];       // A tiles, row pad 4 -> conflict-free b64 reads
    __shared__ float stage[8][32][17];     // per-wave h staging (2 tiles = 32 token rows)
    __shared__ float lval[8][32][33];      // per-wave top-32 values (33-pad, conflict-free)
    __shared__ int   lidx[8][32][33];
    __shared__ float fval[32][TOPK];
    __shared__ int   fidx[32][TOPK];
    __shared__ float s_mean[32], s_norm[32];

    const int tid  = threadIdx.x;
    const int wave = tid >> 5;
    const int lane = tid & 31;
    const int tok0 = blockIdx.x * 32;

    // ---- stage A (32 tokens x 1024 features) into padded LDS, float4 ----
    {
        const float4* src = (const float4*)(xc + (size_t)tok0 * D_MODEL);
        for (int i = tid; i < 32 * (D_MODEL / 4); i += 256) {
            const int t  = i >> 8;            // token 0..31
            const int c4 = i & 255;           // float4 column
            float4 v = src[t * (D_MODEL / 4) + c4];
            *(float4*)&Alds[t][c4 * 4] = v;
        }
    }
    if (tid < 32) { s_mean[tid] = meanArr[tok0 + tid]; s_norm[tid] = normArr[tok0 + tid]; }
    for (int i = tid; i < 8 * 32 * TOPK; i += 256) {
        const int w = i >> 10;
        const int r = i & 1023;
        const int t = r >> 5;
        const int s = r & 31;
        lval[w][t][s] = NEG_INF; lidx[w][t][s] = 0;
    }
    __syncthreads();

    // ---- encode + running top-k ----
    float curmin = NEG_INF;                  // min of this lane's token list
    const int m    = lane & 15;              // WMMA row (A) / col (B) index
    const int ksub = (lane >> 4) * 2;        // K phase (lanes 16-31 hold K+2,K+3)
    const float* arow0 = &Alds[m][ksub];       // tokens  0..15
    const float* arow1 = &Alds[16 + m][ksub];  // tokens 16..31

    for (int tile = wave; tile < N_LAT / 16; tile += 8) {
        const int nbase = tile * 16;
        const int nlat  = nbase + m;
        const float* bp = W_enc + (size_t)nlat * D_MODEL + ksub;
        if (tile + 8 < N_LAT / 16)           // warm L2/L0 for this wave's next tile
            __builtin_prefetch(W_enc + (size_t)(nlat + 128) * D_MODEL + ksub, 0, 0);

        v8f c0 = {};
        v8f c1 = {};
#pragma unroll 4
        for (int k = 0; k < D_MODEL; k += 4) {
            v2f b  = *(const v2f*)(bp + k);              // B = W_enc^T slice (shared)
            v2f a0 = *(const v2f*)(arow0 + k);
            v2f a1 = *(const v2f*)(arow1 + k);
            c0 = __builtin_amdgcn_wmma_f32_16x16x4_f32(
                     false, a0, false, b, (short)0, c0, false, false);
            c1 = __builtin_amdgcn_wmma_f32_16x16x4_f32(
                     false, a1, false, b, (short)0, c1, false, false);
        }
        const float be = b_enc[nlat];
#pragma unroll
        for (int r = 0; r < 8; ++r) {                    // C layout: M = r + 8*(lane>=16)
            stage[wave][r + (lane >> 4) * 8][m]      = c0[r] + be;
            stage[wave][16 + r + (lane >> 4) * 8][m] = c1[r] + be;
        }
        asm volatile("s_wait_dscnt 0" ::: "memory");     // intra-wave LDS RAW fence

        {                                                // lane t scans token t's 16 h values
            float* Lv = &lval[wave][lane][0];
            int*   Li = &lidx[wave][lane][0];
            float cmin = curmin;
            for (int j = 0; j < 16; ++j) {
                const float v = stage[wave][lane][j];
                if (v > cmin) {                          // rare: replace current min entry
                    int slot = 0; float mv = Lv[0];
#pragma unroll
                    for (int s = 1; s < TOPK; ++s) { const float lv = Lv[s]; if (lv < mv) { mv = lv; slot = s; } }
                    Lv[slot] = v; Li[slot] = nbase + j;
                    cmin = v;
#pragma unroll
                    for (int s = 0; s < TOPK; ++s) cmin = fminf(cmin, Lv[s]);
                }
            }
            curmin = cmin;
        }
        asm volatile("" ::: "memory");
    }
    __syncthreads();

    // ---- merge 8 per-wave lists -> global top-32 per token ----
    if (tid < 32) {
        const int t = tid;
        for (int k = 0; k < TOPK; ++k) {
            float best = NEG_INF; int bw = 0, bs = 0;
            for (int w = 0; w < 8; ++w)
                for (int s = 0; s < TOPK; ++s) {
                    const float v = lval[w][t][s];
                    if (v > best) { best = v; bw = w; bs = s; }
                }
            fval[t][k] = best;
            fidx[t][k] = lidx[bw][t][bs];
            lval[bw][t][bs] = NEG_INF;
        }
    }
    __syncthreads();

    // ---- sparse decode + denormalize ----
    for (int t = 0; t < 32; ++t) {
        const float nrm = s_norm[t], mn = s_mean[t];
        for (int d = tid; d < D_MODEL; d += 256) {
            float acc = b_pre[d];
#pragma unroll
            for (int k = 0; k < TOPK; ++k) {
                const float v = fval[t][k];              // uniform across threads
                if (v > 0.f)                              // ReLU on top-k values
                    acc = fmaf(v, WdT[(size_t)fidx[t][k] * D_MODEL + d], acc);
            }
            out[(size_t)(tok0 + t) * D_MODEL + d] = acc * nrm + mn;
        }
    }
}

// -------------------------------------------------------------------------
extern "C" void kernel_launch(void* const* d_in, const int* in_sizes, int n_in,
                              void* d_out, int out_size, void* d_ws, size_t ws_size,
                              hipStream_t stream)
{
    const float* x     = (const float*)d_in[0];   // [2,2048,1024]
    const float* b_pre = (const float*)d_in[1];   // [1024]
    const float* Wenc  = (const float*)d_in[2];   // [16384,1024]
    const float* b_enc = (const float*)d_in[3];   // [16384]
    const float* Wdec  = (const float*)d_in[4];   // [1024,16384]
    float* out = (float*)d_out;

    const int N = in_sizes[0] / D_MODEL;          // 4096 tokens

    // workspace layout: xc [N,1024] | WdT [16384,1024] | mean [N] | norm [N]
    float* xc   = (float*)d_ws;
    float* WdT  = xc  + (size_t)N * D_MODEL;
    float* mean = WdT + (size_t)N_LAT * D_MODEL;
    float* norm = mean + N;

    prep_kernel<<<N, 256, 0, stream>>>(x, b_pre, xc, mean, norm);
    transpose_kernel<<<dim3(N_LAT / 32, D_MODEL / 32), dim3(32, 8), 0, stream>>>(Wdec, WdT);
    fused_kernel<<<N / 32, 256, 0, stream>>>(xc, Wenc, b_enc, b_pre, WdT, mean, norm, out);
}